// NonSOCSymmetricContraction_85332410237129
// MI455X (gfx1250) — compile-verified
//
#include <hip/hip_runtime.h>
#include <hip/hip_bf16.h>

typedef float v2f __attribute__((ext_vector_type(2)));
typedef float v8f __attribute__((ext_vector_type(8)));

#define NB 4096
#define NA 64
#define NL 16
#define NM 4
#define NE 10
#define P_TOT 416           // 32 + 128 + 256 unified path index
#define P1_END 32
#define P2_END 160
#define EPAD 16             // e padded to 16 for WMMA N
#define XPITCH 68           // LDS pitch for 64-float x rows (conflict-free gathers)

// ---------------- prep 1: packed gather offsets per path p ----------------
// packed = off1 | off2<<8 | off3<<16 ; off = i*4 + l into x[b,a,:,:]
__global__ void prep_offsets_kernel(const int* __restrict__ i1, const int* __restrict__ l1,
                                    const int* __restrict__ i2, const int* __restrict__ j2,
                                    const int* __restrict__ l2, const int* __restrict__ m2,
                                    const int* __restrict__ i3, const int* __restrict__ j3,
                                    const int* __restrict__ f3, const int* __restrict__ l3,
                                    const int* __restrict__ m3, const int* __restrict__ g3,
                                    int* __restrict__ offs) {
  int p = blockIdx.x * blockDim.x + threadIdx.x;
  if (p >= P_TOT) return;
  int packed;
  if (p < P1_END) {
    int o1 = i1[p] * NM + l1[p];
    packed = o1;
  } else if (p < P2_END) {
    int q = p - P1_END;
    int o1 = i2[q] * NM + l2[q];
    int o2 = j2[q] * NM + m2[q];
    packed = o1 | (o2 << 8);
  } else {
    int q = p - P2_END;
    int o1 = i3[q] * NM + l3[q];
    int o2 = j3[q] * NM + m3[q];
    int o3 = f3[q] * NM + g3[q];
    packed = o1 | (o2 << 8) | (o3 << 16);
  }
  offs[p] = packed;
}

// ---------------- prep 2: gathered+coeff-scaled weights -------------------
// Wg[a][p][n] = (n < 10) ? coeff[p] * w_path(p)[n, k[p], q[p], a] : 0
// one thread per (a,p): 64*416 = 26624 = 104 blocks * 256
__global__ void prep_wg_kernel(const float* __restrict__ w1, const float* __restrict__ w2,
                               const float* __restrict__ w3,
                               const float* __restrict__ c1, const float* __restrict__ c2,
                               const float* __restrict__ c3,
                               const int* __restrict__ k1, const int* __restrict__ q1,
                               const int* __restrict__ k2, const int* __restrict__ q2,
                               const int* __restrict__ k3, const int* __restrict__ q3,
                               float* __restrict__ Wg) {
  int tid = blockIdx.x * blockDim.x + threadIdx.x;
  if (tid >= NA * P_TOT) return;
  int p = tid % P_TOT;
  int a = tid / P_TOT;
  float cf;
  const float* wbase;
  int estride;   // flat = e*estride + (k*Q + q)*NA + a
  int kq;
  if (p < P1_END) {
    cf = c1[p];
    wbase = w1; kq = k1[p] * 2 + q1[p]; estride = 4 * 2 * NA;
  } else if (p < P2_END) {
    int t = p - P1_END;
    cf = c2[t];
    wbase = w2; kq = k2[t] * 4 + q2[t]; estride = 16 * 4 * NA;
  } else {
    int t = p - P2_END;
    cf = c3[t];
    wbase = w3; kq = k3[t] * 6 + q3[t]; estride = 32 * 6 * NA;
  }
  const int kqstride = kq * NA + a;
  float* dst = Wg + ((size_t)a * P_TOT + p) * EPAD;
#pragma unroll
  for (int n = 0; n < EPAD; ++n) {
    dst[n] = (n < NE) ? cf * wbase[(size_t)n * estride + kqstride] : 0.0f;
  }
}

// one K=4 WMMA step; CNT = number of x factors (compile-time per segment)
template <int CNT>
__device__ __forceinline__ void wmma_step(int t, const int* __restrict__ ldsOff,
                                          const float* __restrict__ X,
                                          const float* __restrict__ ldsW,
                                          int half, int lrow, v8f& acc) {
  const int pA = 4 * t + 2 * half;            // this lane's K-slots: pA, pA+1
  const int pk0 = ldsOff[pA];
  const int pk1 = ldsOff[pA + 1];
  float xa = X[pk0 & 0xFF];
  float xb = X[pk1 & 0xFF];
  if (CNT > 1) { xa *= X[(pk0 >> 8) & 0xFF]; xb *= X[(pk1 >> 8) & 0xFF]; }
  if (CNT > 2) { xa *= X[(pk0 >> 16) & 0xFF]; xb *= X[(pk1 >> 16) & 0xFF]; }
  v2f af; af.x = xa; af.y = xb;
  v2f bf;
  bf.x = ldsW[pA * EPAD + lrow];
  bf.y = ldsW[(pA + 1) * EPAD + lrow];
  acc = __builtin_amdgcn_wmma_f32_16x16x4_f32(false, af, false, bf,
                                              (short)0, acc, false, false);
}

// ---------------- main fused kernel ----------------------------------------
// grid (32, 8): x = 128-b tile (8 waves x 16 b), y = 8-a group; block = 256 (wave32 x 8)
__global__ __launch_bounds__(256) void contract_kernel(const float* __restrict__ x,
                                                       const float* __restrict__ y,
                                                       const float* __restrict__ Wg,
                                                       const int* __restrict__ offs,
                                                       float* __restrict__ out) {
  __shared__ float ldsW[P_TOT * EPAD];        // 26,624 B : Wg[a] slice, shared by all waves
  __shared__ float ldsX[8 * 16 * XPITCH];     // 34,816 B : per-wave 16 x-rows, pitch 68
  __shared__ float ldsY[128 * EPAD];          //  8,192 B : y tile (e padded w/ zeros)
  __shared__ int   ldsOff[P_TOT];             //  1,664 B

  const int tid  = threadIdx.x;
  const int wv   = tid >> 5;
  const int lane = tid & 31;
  const int half = lane >> 4;
  const int lrow = lane & 15;
  const int b0   = blockIdx.x * 128;
  const int a0   = blockIdx.y * 8;
  const int myb  = b0 + wv * 16;

  for (int i = tid; i < P_TOT; i += 256) ldsOff[i] = offs[i];
  for (int i = tid; i < 128 * EPAD; i += 256) {
    int row = i >> 4, e = i & 15;
    ldsY[i] = (e < NE) ? y[(size_t)(b0 + row) * NE + e] : 0.0f;
  }

  for (int ai = 0; ai < 8; ++ai) {
    const int a = a0 + ai;
    __syncthreads();
    // stage Wg[a] (416x16 f32) cooperatively
    {
      const float4* src = (const float4*)(Wg + (size_t)a * P_TOT * EPAD);
      float4* dst = (float4*)ldsW;
      for (int i = tid; i < P_TOT * EPAD / 4; i += 256) dst[i] = src[i];
    }
    // stage this wave's 16 x rows: x[myb+row, a, 0:64]
    for (int i = lane; i < 16 * 16; i += 32) {
      int row = i >> 4, c4 = i & 15;
      const float4 v = *(const float4*)(x + ((size_t)(myb + row) * NA + a) * 64 + c4 * 4);
      *(float4*)(&ldsX[(wv * 16 + row) * XPITCH + c4 * 4]) = v;
    }
    __syncthreads();

    const float* X = &ldsX[(wv * 16 + lrow) * XPITCH];
    v8f acc = {};
    // K = 416 contraction: H[b,e] = sum_p xprod[b,p] * Wg[p,e]
    // Segmented by path so the factor count is compile-time: no divergent
    // branches, EXEC stays all-ones, gathers batch per unrolled group.
#pragma unroll
    for (int t = 0; t < 8; ++t)            // path 1: p in [0,32), 1 factor
      wmma_step<1>(t, ldsOff, X, ldsW, half, lrow, acc);
#pragma unroll 4
    for (int t = 8; t < 40; ++t)           // path 2: p in [32,160), 2 factors
      wmma_step<2>(t, ldsOff, X, ldsW, half, lrow, acc);
#pragma unroll 4
    for (int t = 40; t < 104; ++t)         // path 3: p in [160,416), 3 factors
      wmma_step<3>(t, ldsOff, X, ldsW, half, lrow, acc);

    // out[b,a] = sum_e y[b,e] * H[b,e] ; C/D layout: vgpr r -> rows r / r+8
#pragma unroll
    for (int r = 0; r < 8; ++r) {
      const int row = wv * 16 + r + half * 8;
      float pr = acc[r] * ldsY[row * EPAD + lrow];
      pr += __shfl_xor(pr, 1, 16);
      pr += __shfl_xor(pr, 2, 16);
      pr += __shfl_xor(pr, 4, 16);
      pr += __shfl_xor(pr, 8, 16);
      if (lrow == 0) out[(size_t)(b0 + row) * NA + a] = pr;
    }
  }
}

extern "C" void kernel_launch(void* const* d_in, const int* in_sizes, int n_in,
                              void* d_out, int out_size, void* d_ws, size_t ws_size,
                              hipStream_t stream) {
  const float* x  = (const float*)d_in[0];
  const float* y  = (const float*)d_in[1];
  const float* w1 = (const float*)d_in[2];
  const float* w2 = (const float*)d_in[3];
  const float* w3 = (const float*)d_in[4];
  const float* c1 = (const float*)d_in[5];
  const float* c2 = (const float*)d_in[6];
  const float* c3 = (const float*)d_in[7];
  const int* i1 = (const int*)d_in[8];
  const int* l1 = (const int*)d_in[9];
  const int* k1 = (const int*)d_in[10];
  const int* q1 = (const int*)d_in[11];
  const int* i2 = (const int*)d_in[12];
  const int* j2 = (const int*)d_in[13];
  const int* l2 = (const int*)d_in[14];
  const int* m2 = (const int*)d_in[15];
  const int* k2 = (const int*)d_in[16];
  const int* q2 = (const int*)d_in[17];
  const int* i3 = (const int*)d_in[18];
  const int* j3 = (const int*)d_in[19];
  const int* f3 = (const int*)d_in[20];
  const int* l3 = (const int*)d_in[21];
  const int* m3 = (const int*)d_in[22];
  const int* g3 = (const int*)d_in[23];
  const int* k3 = (const int*)d_in[24];
  const int* q3 = (const int*)d_in[25];

  // workspace layout: [0,1664) packed offsets, [2048, 2048+1,703,936) Wg
  int*   offs = (int*)d_ws;
  float* Wg   = (float*)((char*)d_ws + 2048);
  float* out  = (float*)d_out;

  prep_offsets_kernel<<<2, 256, 0, stream>>>(i1, l1, i2, j2, l2, m2,
                                             i3, j3, f3, l3, m3, g3, offs);
  prep_wg_kernel<<<104, 256, 0, stream>>>(w1, w2, w3, c1, c2, c3,
                                          k1, q1, k2, q2, k3, q3, Wg);
  contract_kernel<<<dim3(32, 8), 256, 0, stream>>>(x, y, Wg, offs, out);
}